// NeuralTrustNetwork_13503377179004
// MI455X (gfx1250) — compile-verified
//
#include <hip/hip_runtime.h>
#include <hip/hip_bf16.h>

typedef float v2f __attribute__((ext_vector_type(2)));
typedef float v8f __attribute__((ext_vector_type(8)));

#define D 64
#define WPAD 68   // padded transposed-row stride: conflict-free b64 B-fragment loads
#define WAVES_PER_BLOCK 8
#define EDGES_PER_WAVE 16
#define EDGES_PER_BLOCK (WAVES_PER_BLOCK * EDGES_PER_WAVE)

__device__ __forceinline__ float lrelu(float v) {
    // negative_slope = 0.01: max(v, 0.01v) is exact for both signs
    return fmaxf(v, 0.01f * v);
}

// s = a + b elementwise over node tables, float4-vectorized
__global__ void node_sum_kernel(const float4* __restrict__ a,
                                const float4* __restrict__ b,
                                float4* __restrict__ o, int n4) {
    int i = blockIdx.x * blockDim.x + threadIdx.x;
    if (i < n4) {
        float4 va = a[i], vb = b[i], r;
        r.x = va.x + vb.x; r.y = va.y + vb.y;
        r.z = va.z + vb.z; r.w = va.w + vb.w;
        o[i] = r;
    }
}

// One wave32 handles a 16-edge tile.
// A-fragment (16x4 fp32, ISA layout): lane = (r = lane&15, h = lane>>4),
//   VGPR v holds A[r][4*kc + 2*h + v].
// B-fragment (4x16 fp32, mirrored):  lane = (n = lane&15, h),
//   VGPR v holds B[2*h + v][n]  -> contiguous pair in transposed LDS copy.
// C/D tile (16x16 fp32, ISA layout): lane (n, h), VGPR v holds C[v + 8*h][n].
template <bool FUSED>
__global__ __launch_bounds__(256)
void trust_edge_kernel(const int* __restrict__ src, const int* __restrict__ dst,
                       const float* __restrict__ sA, const float* __restrict__ sB,
                       const float* __restrict__ pA, const float* __restrict__ pB,
                       const float* __restrict__ x,  const float* __restrict__ w,
                       const float* __restrict__ Wmlp, const float* __restrict__ bmlp,
                       const float* __restrict__ wL, const float* __restrict__ bL,
                       const float* __restrict__ w1, const float* __restrict__ b1,
                       const float* __restrict__ w2, const float* __restrict__ b2,
                       float* __restrict__ out, int E)
{
    __shared__ float lWt[D * WPAD];          // W_mlp transposed: lWt[col*WPAD + k]
    __shared__ float lbm[D], lwL[D], lw1[D], lw2[D];
    __shared__ float lbias;

    // Stage W_mlp transposed (coalesced global reads, scattered LDS writes,
    // once per block) + small vectors into LDS.
    {
        for (int i = threadIdx.x; i < D * D; i += blockDim.x) {
            const int k = i >> 6, col = i & (D - 1);
            lWt[col * WPAD + k] = Wmlp[i];
        }
        int t = threadIdx.x;
        if (t < D) { lbm[t] = bmlp[t]; lwL[t] = wL[t]; lw1[t] = w1[t]; lw2[t] = w2[t]; }
        if (t == 0) lbias = bL[0] + b1[0] + b2[0];
    }
    __syncthreads();

    const int lane = threadIdx.x & 31;
    const int wave = threadIdx.x >> 5;
    const int n = lane & 15;
    const int h = lane >> 4;
    const int base = (blockIdx.x * WAVES_PER_BLOCK + wave) * EDGES_PER_WAVE;

    int e = base + n;
    if (e >= E) e = E - 1;               // clamp loads; stores guarded below
    const int si = src[e], di = dst[e];

    const float* xs  = x  + (size_t)si * D;
    const float* xd  = x  + (size_t)di * D;
    const float* wsr = w  + (size_t)si * D;
    const float* wdr = w  + (size_t)di * D;
    const float* sr0 = sA + (size_t)si * D;
    const float* pr0 = pA + (size_t)di * D;
    const float* sr1 = FUSED ? (sB + (size_t)si * D) : nullptr;
    const float* pr1 = FUSED ? (pB + (size_t)di * D) : nullptr;

    // Gather A fragments (lrelu(c)) + accumulate the two u_mul_v dot heads.
    v2f a[16];
    float dot = 0.f;
#pragma unroll
    for (int kc = 0; kc < 16; ++kc) {
        const int f = 4 * kc + 2 * h;
        float2 sv = *(const float2*)(sr0 + f);
        float2 pv = *(const float2*)(pr0 + f);
        float c0 = sv.x + pv.x, c1 = sv.y + pv.y;
        if (FUSED) {
            float2 sv1 = *(const float2*)(sr1 + f);
            float2 pv1 = *(const float2*)(pr1 + f);
            c0 += sv1.x + pv1.x;
            c1 += sv1.y + pv1.y;
        }
        a[kc][0] = lrelu(c0);
        a[kc][1] = lrelu(c1);

        float2 xv0 = *(const float2*)(xs + f);
        float2 xv1 = *(const float2*)(xd + f);
        float2 wv0 = *(const float2*)(wsr + f);
        float2 wv1 = *(const float2*)(wdr + f);
        dot += xv0.x * xv1.x * lw1[f] + xv0.y * xv1.y * lw1[f + 1];
        dot += wv0.x * wv1.x * lw2[f] + wv0.y * wv1.y * lw2[f + 1];
    }

    // [16,64] x [64,64] fp32 GEMM on the matrix pipe: 4 N-tiles x 16 K-chunks.
    float t8[8] = {0.f, 0.f, 0.f, 0.f, 0.f, 0.f, 0.f, 0.f};
#pragma unroll
    for (int nt = 0; nt < 4; ++nt) {
        v8f acc = {0.f, 0.f, 0.f, 0.f, 0.f, 0.f, 0.f, 0.f};
        const int col = nt * 16 + n;
        const float* brow = &lWt[col * WPAD + 2 * h];
#pragma unroll
        for (int kc = 0; kc < 16; ++kc) {
            // single aligned ds_load_b64: B[2h..2h+1][col] of K-chunk kc
            v2f b = *(const v2f*)(brow + 4 * kc);
            acc = __builtin_amdgcn_wmma_f32_16x16x4_f32(
                /*neg_a=*/false, a[kc], /*neg_b=*/false, b,
                /*c_mod=*/(short)0, acc, /*reuse_a=*/false, /*reuse_b=*/false);
        }
        const float bm = lbm[col];
        const float wl = lwL[col];
#pragma unroll
        for (int v = 0; v < 8; ++v)
            t8[v] += lrelu(acc[v] + bm) * wl;   // h_L @ wL partials, per lane
    }

    // dot for edge r lives split across lanes (r, r+16): merge halves.
    float dfull = dot + __shfl_xor(dot, 16, 32);

#pragma unroll
    for (int v = 0; v < 8; ++v) {
        // reduce over N (16 lanes per half); masks <16 stay within each half
        float val = t8[v];
        val += __shfl_xor(val, 1, 32);
        val += __shfl_xor(val, 2, 32);
        val += __shfl_xor(val, 4, 32);
        val += __shfl_xor(val, 8, 32);
        // edge for this (v,h) is m = v + 8h; its dot head sits in lane v + 8h
        float dm = __shfl(dfull, v + 8 * h, 32);
        if (n == v) {
            int m = base + v + 8 * h;
            if (m < E) out[m] = val + dm + lbias;
        }
    }
}

extern "C" void kernel_launch(void* const* d_in, const int* in_sizes, int n_in,
                              void* d_out, int out_size, void* d_ws, size_t ws_size,
                              hipStream_t stream) {
    const int*   src  = (const int*)d_in[0];
    const int*   dst  = (const int*)d_in[1];
    const float* s1   = (const float*)d_in[2];
    const float* s2   = (const float*)d_in[3];
    const float* p1   = (const float*)d_in[4];
    const float* p2   = (const float*)d_in[5];
    const float* x    = (const float*)d_in[6];
    const float* w    = (const float*)d_in[7];
    const float* Wmlp = (const float*)d_in[8];
    const float* bmlp = (const float*)d_in[9];
    const float* wL   = (const float*)d_in[10];
    const float* bL   = (const float*)d_in[11];
    const float* w1   = (const float*)d_in[12];
    const float* b1   = (const float*)d_in[13];
    const float* w2   = (const float*)d_in[14];
    const float* b2   = (const float*)d_in[15];
    float* out = (float*)d_out;

    const int E = in_sizes[0];
    const size_t nodeFloats = (size_t)in_sizes[2];       // N * D
    const int blocks = (E + EDGES_PER_BLOCK - 1) / EDGES_PER_BLOCK;

    const size_t need = 2 * nodeFloats * sizeof(float);
    if (ws_size >= need && (nodeFloats % 4) == 0) {
        float* sbuf = (float*)d_ws;
        float* pbuf = sbuf + nodeFloats;
        const int n4 = (int)(nodeFloats / 4);
        const int pb = (n4 + 255) / 256;
        node_sum_kernel<<<pb, 256, 0, stream>>>((const float4*)s1, (const float4*)s2,
                                                (float4*)sbuf, n4);
        node_sum_kernel<<<pb, 256, 0, stream>>>((const float4*)p1, (const float4*)p2,
                                                (float4*)pbuf, n4);
        trust_edge_kernel<false><<<blocks, 256, 0, stream>>>(
            src, dst, sbuf, nullptr, pbuf, nullptr, x, w,
            Wmlp, bmlp, wL, bL, w1, b1, w2, b2, out, E);
    } else {
        trust_edge_kernel<true><<<blocks, 256, 0, stream>>>(
            src, dst, s1, s2, p1, p2, x, w,
            Wmlp, bmlp, wL, bL, w1, b1, w2, b2, out, E);
    }
}